// SpectralGaussianRenderer_72499047957325
// MI455X (gfx1250) — compile-verified
//
#include <hip/hip_runtime.h>
#include <hip/hip_bf16.h>

#define N_G   512
#define H_IMG 324
#define W_IMG 332
#define C_SPEC 120
#define C_PAD  128
#define SPEC_STRIDE 132          // LDS row stride (dwords) to avoid bank conflicts
#define P_IMG (H_IMG * W_IMG)

typedef __attribute__((ext_vector_type(2))) float v2f;
typedef __attribute__((ext_vector_type(8))) float v8f;

// ---- workspace layout (float offsets) ----
#define WS_SX   0
#define WS_SY   512
#define WS_C00  1024
#define WS_C11  1536
#define WS_OP   2048            // effective opacity (0 if !valid or off-screen)
#define WS_DEP  2560
#define WS_MA   3072            // per-step sum of alpha over all pixels
#define WS_SPEC 3584            // sorted spec, [512][C_PAD]
#define WS_NSTAR (3584 + N_G * C_PAD)   // int slot

// ---------------------------------------------------------------------------
// K1: project, cull, diagonal cov-inverse, stable depth-rank sort (O(N^2)),
//     scatter everything into sorted order, zero the meanAlpha array.
// ---------------------------------------------------------------------------
__global__ __launch_bounds__(N_G) void prep_kernel(
    const float* __restrict__ pos, const float* __restrict__ scales,
    const float* __restrict__ ops, const float* __restrict__ spec,
    const float* __restrict__ intr, const float* __restrict__ extr,
    float* __restrict__ ws)
{
    __shared__ float sdep[N_G];
    const int i = threadIdx.x;

    const float px = pos[i * 3 + 0], py = pos[i * 3 + 1], pz = pos[i * 3 + 2];
    float cam[3];
#pragma unroll
    for (int j = 0; j < 3; ++j)
        cam[j] = extr[j * 4 + 0] * px + extr[j * 4 + 1] * py +
                 extr[j * 4 + 2] * pz + extr[j * 4 + 3];
    const float depth = cam[2];
    const float pr0 = intr[0] * cam[0] + intr[1] * cam[1] + intr[2] * cam[2];
    const float pr1 = intr[3] * cam[0] + intr[4] * cam[1] + intr[5] * cam[2];
    const float pr2 = intr[6] * cam[0] + intr[7] * cam[1] + intr[8] * cam[2];
    const float sx = pr0 / (pr2 + 1e-6f);
    const float sy = pr1 / (pr2 + 1e-6f);

    bool valid = (depth > 0.01f) && (depth < 100.0f) &&
                 (sx > -100.0f) && (sx < (float)W_IMG + 100.0f) &&
                 (sy > -100.0f) && (sy < (float)H_IMG + 100.0f);
    bool off = (sx < -(float)W_IMG) || (sx > 2.0f * (float)W_IMG) ||
               (sy < -(float)H_IMG) || (sy > 2.0f * (float)H_IMG);
    const float opEff = (valid && !off) ? ops[i] : 0.0f;

    const float s0 = scales[i * 3 + 0], s1 = scales[i * 3 + 1];
    const float c00 = 1.0f / (s0 * s0 + 1e-4f);
    const float c11 = 1.0f / (s1 * s1 + 1e-4f);

    sdep[i] = depth;
    __syncthreads();
    int rank = 0;
    for (int j = 0; j < N_G; ++j) {
        const float dj = sdep[j];
        rank += (dj < depth) || (dj == depth && j < i);   // stable argsort rank
    }

    ws[WS_SX  + rank] = sx;
    ws[WS_SY  + rank] = sy;
    ws[WS_C00 + rank] = c00;
    ws[WS_C11 + rank] = c11;
    ws[WS_OP  + rank] = opEff;
    ws[WS_DEP + rank] = depth;
    ws[WS_MA  + i] = 0.0f;
    for (int c = 0; c < C_SPEC; ++c)
        ws[WS_SPEC + rank * C_PAD + c] = spec[i * C_SPEC + c];
    for (int c = C_SPEC; c < C_PAD; ++c)
        ws[WS_SPEC + rank * C_PAD + c] = 0.0f;
}

// ---------------------------------------------------------------------------
// K2: phase-1 scan — per-pixel A trajectory (no done-skip), accumulate
//     Sum_p alpha_n(p) into ws[WS_MA + n] via wave+LDS reduction + atomicAdd.
//     Tile: 16x16 pixels, 256 threads.
// ---------------------------------------------------------------------------
__global__ __launch_bounds__(256) void ascan_kernel(float* __restrict__ ws)
{
    __shared__ float gsx[N_G], gsy[N_G], gc00[N_G], gc11[N_G], gop[N_G];
    __shared__ float part[8];
    const int t = threadIdx.x;
    for (int k = t; k < N_G; k += 256) {
        gsx[k] = ws[WS_SX + k];  gsy[k] = ws[WS_SY + k];
        gc00[k] = ws[WS_C00 + k]; gc11[k] = ws[WS_C11 + k];
        gop[k] = ws[WS_OP + k];
    }
    __syncthreads();

    const int x = blockIdx.x * 16 + (t & 15);
    const int y = blockIdx.y * 16 + (t >> 4);
    const float fx = (float)x, fy = (float)y;
    const float inb = (x < W_IMG && y < H_IMG) ? 1.0f : 0.0f;
    const int wv = t >> 5, ln = t & 31;

    float A = 0.0f;
    for (int n = 0; n < N_G; ++n) {
        const float dx = fx - gsx[n], dy = fy - gsy[n];
        const float m = dx * dx * gc00[n] + dy * dy * gc11[n];
        float a = gop[n] * __expf(-0.5f * m) * (1.0f - A) * inb;
        A += a;
        // wave32 reduce
        float s = a;
        for (int o = 16; o > 0; o >>= 1) s += __shfl_xor(s, o);
        if (ln == 0) part[wv] = s;
        __syncthreads();
        if (t == 0) {
            float tot = 0.0f;
            for (int w = 0; w < 8; ++w) tot += part[w];
            atomicAdd(&ws[WS_MA + n], tot);
        }
        __syncthreads();
    }
}

// ---------------------------------------------------------------------------
// K3: find n* = first n with mean(A before step n) > 0.99
// ---------------------------------------------------------------------------
__global__ void nstar_kernel(float* __restrict__ ws)
{
    if (threadIdx.x == 0 && blockIdx.x == 0) {
        const float thr = 0.99f * (float)P_IMG;
        float cum = 0.0f;
        int ns = N_G;
        for (int n = 0; n < N_G; ++n) {
            if (cum > thr) { ns = n; break; }
            cum += ws[WS_MA + n];
        }
        ((int*)ws)[WS_NSTAR] = ns;
    }
}

// ---------------------------------------------------------------------------
// K4: phase-2 render. Each wave owns 16 pixels of one row; all 32 lanes
//     redundantly run the scan for pixel (lane&15) so the WMMA B operand
//     (4x16 f32: VGPR0 = {K0 | K2}, VGPR1 = {K1 | K3} across lane halves)
//     needs no cross-lane traffic. Spectral accumulation via
//     V_WMMA_F32_16X16X4_F32, 8 channel tiles (120 padded to 128).
//     Workgroup tile: 16 wide x 8 rows (8 waves, 256 threads).
// ---------------------------------------------------------------------------
__global__ __launch_bounds__(256) void render_kernel(
    const float* __restrict__ ws, const float* __restrict__ tone,
    float* __restrict__ out)
{
    __shared__ float gsx[N_G], gsy[N_G], gc00[N_G], gc11[N_G], gop[N_G], gdep[N_G];
    __shared__ float sspec[16 * SPEC_STRIDE];   // 16-gaussian chunk of spec
    const int t = threadIdx.x;
    for (int k = t; k < N_G; k += 256) {
        gsx[k] = ws[WS_SX + k];   gsy[k] = ws[WS_SY + k];
        gc00[k] = ws[WS_C00 + k]; gc11[k] = ws[WS_C11 + k];
        gop[k] = ws[WS_OP + k];   gdep[k] = ws[WS_DEP + k];
    }
    const int n_star = ((const int*)ws)[WS_NSTAR];
    __syncthreads();

    const int wv = t >> 5, ln = t & 31;
    const int col = ln & 15;        // pixel column (WMMA N index)
    const int kh  = ln >> 4;        // lane half (K split for A/B operands)
    const int x = blockIdx.x * 16 + col;
    const int y = blockIdx.y * 8 + wv;
    const float fx = (float)x, fy = (float)y;
    const float inb = (x < W_IMG && y < H_IMG) ? 1.0f : 0.0f;

    float A = 0.0f, Dacc = 0.0f;
    v8f acc[8] = {};

    for (int nc = 0; nc < N_G; nc += 16) {
        // stage 16 gaussians x 128 channels of sorted spec into LDS
        for (int q = t; q < 16 * C_PAD; q += 256)
            sspec[(q >> 7) * SPEC_STRIDE + (q & 127)] =
                ws[WS_SPEC + nc * C_PAD + q];
        __syncthreads();

        for (int k4 = 0; k4 < 16; k4 += 4) {
            float al[4];
#pragma unroll
            for (int u = 0; u < 4; ++u) {       // sequential alpha scan
                const int nn = nc + k4 + u;
                const float dx = fx - gsx[nn], dy = fy - gsy[nn];
                const float m = dx * dx * gc00[nn] + dy * dy * gc11[nn];
                float a = gop[nn] * __expf(-0.5f * m) * (1.0f - A) * inb;
                a = (nn < n_star) ? a : 0.0f;
                A += a;
                Dacc += gdep[nn] * a;
                al[u] = a;
            }
            v2f B;
            B.x = kh ? al[2] : al[0];
            B.y = kh ? al[3] : al[1];
#pragma unroll
            for (int ct = 0; ct < 8; ++ct) {
                v2f Aop;
                const int base = (k4 + 2 * kh) * SPEC_STRIDE + ct * 16 + col;
                Aop.x = sspec[base];
                Aop.y = sspec[base + SPEC_STRIDE];
                acc[ct] = __builtin_amdgcn_wmma_f32_16x16x4_f32(
                    false, Aop, false, B, (short)0, acc[ct], false, false);
            }
        }
        __syncthreads();
    }

    if (inb != 0.0f) {
        const int p = y * W_IMG + x;
        if (ln < 16) {
            out[C_SPEC * P_IMG + p] = Dacc;                // depth image
            out[C_SPEC * P_IMG + P_IMG + p] = A;           // A_final
        }
        const float oma = 1.0f - A;                        // background term
#pragma unroll
        for (int ct = 0; ct < 8; ++ct) {
#pragma unroll
            for (int j = 0; j < 8; ++j) {
                const int c = ct * 16 + j + 8 * kh;        // C/D lane layout
                if (c < C_SPEC)
                    out[c * P_IMG + p] = (acc[ct][j] + oma) * tone[c];
            }
        }
    }
}

// ---------------------------------------------------------------------------
extern "C" void kernel_launch(void* const* d_in, const int* in_sizes, int n_in,
                              void* d_out, int out_size, void* d_ws, size_t ws_size,
                              hipStream_t stream)
{
    const float* pos    = (const float*)d_in[0];
    /* d_in[1] = rotations (unused by reference) */
    const float* scales = (const float*)d_in[2];
    const float* ops    = (const float*)d_in[3];
    const float* spec   = (const float*)d_in[4];
    const float* tone   = (const float*)d_in[5];
    const float* intr   = (const float*)d_in[6];
    const float* extr   = (const float*)d_in[7];
    float* ws  = (float*)d_ws;
    float* out = (float*)d_out;

    prep_kernel<<<1, N_G, 0, stream>>>(pos, scales, ops, spec, intr, extr, ws);

    dim3 g2((W_IMG + 15) / 16, (H_IMG + 15) / 16);
    ascan_kernel<<<g2, 256, 0, stream>>>(ws);

    nstar_kernel<<<1, 32, 0, stream>>>(ws);

    dim3 g4((W_IMG + 15) / 16, (H_IMG + 7) / 8);
    render_kernel<<<g4, 256, 0, stream>>>(ws, tone, out);
}